// Attention_52312701665770
// MI455X (gfx1250) — compile-verified
//
#include <hip/hip_runtime.h>
#include <hip/hip_bf16.h>
#include <math.h>

// Problem sizes (fixed by the reference)
#define B_ 32
#define L_ 2048
#define D_ 1024
#define C_ 1024
#define LC 128            // L-chunk per workgroup in the streaming pass
#define NC (L_ / LC)      // 16 chunks per batch row

typedef float v2f __attribute__((ext_vector_type(2)));
typedef float v8f __attribute__((ext_vector_type(8)));

// Workspace layout (floats). Total = 623616 floats (~2.4 MB).
constexpr int WS_V   = 0;                  // v[b,d] = query @ W_h        : B*D
constexpr int WS_E   = WS_V + B_ * D_;     // masked energies             : B*L
constexpr int WS_M   = WS_E + B_ * L_;     // per-chunk max               : B*NC
constexpr int WS_S   = WS_M + B_ * NC;     // per-chunk exp-sum           : B*NC
constexpr int WS_CTX = WS_S + B_ * NC;     // per-chunk partial context   : B*NC*D

// ---------------------------------------------------------------------------
// K1: v[b,d] = sum_c query[b,c] * W_h[c,d]    (32x1024 @ 1024x1024, f32 WMMA)
// One wave per 16x16 output tile; K stepped by 4 via V_WMMA_F32_16X16X4_F32.
// A (16x4 f32) layout: lane = (K/2)*16 + M, vgpr = K%2.  B mirrors with N.
// C/D (16x16 f32) layout: vgpr r, lane l -> M = r + (l/16)*8, N = l%16.
// ---------------------------------------------------------------------------
__global__ __launch_bounds__(32) void k_qw_gemm(const float* __restrict__ q,
                                                const float* __restrict__ W,
                                                float* __restrict__ v) {
  const int lane = threadIdx.x & 31;
  const int mt = blockIdx.x & 1;   // 2 row tiles (B=32)
  const int nt = blockIdx.x >> 1;  // 64 col tiles (D=1024)
  const int m0 = mt * 16, n0 = nt * 16;
  const int mrow  = m0 + (lane & 15);
  const int ncol  = n0 + (lane & 15);
  const int khalf = (lane >> 4) * 2;  // lanes 16..31 hold K+2,K+3

#if __has_builtin(__builtin_amdgcn_wmma_f32_16x16x4_f32)
  v8f acc = {};
  #pragma unroll 4
  for (int k0 = 0; k0 < C_; k0 += 4) {
    const int kb = k0 + khalf;
    v2f a;
    a[0] = q[(size_t)mrow * C_ + kb];
    a[1] = q[(size_t)mrow * C_ + kb + 1];
    v2f bm;
    bm[0] = W[(size_t)kb * D_ + ncol];
    bm[1] = W[(size_t)(kb + 1) * D_ + ncol];
    acc = __builtin_amdgcn_wmma_f32_16x16x4_f32(
        /*neg_a=*/false, a, /*neg_b=*/false, bm,
        /*c_mod=*/(short)0, acc, /*reuse_a=*/false, /*reuse_b=*/false);
  }
  #pragma unroll
  for (int r = 0; r < 8; ++r) {
    const int m = m0 + r + (lane >> 4) * 8;
    v[(size_t)m * D_ + n0 + (lane & 15)] = acc[r];
  }
#else
  // Scalar fallback (kept so compile never fails; histogram reveals which path).
  #pragma unroll
  for (int r = 0; r < 8; ++r) {
    const int m = m0 + r + (lane >> 4) * 8;
    const int n = n0 + (lane & 15);
    float s = 0.f;
    for (int k = 0; k < C_; ++k) s = fmaf(q[(size_t)m * C_ + k], W[(size_t)k * D_ + n], s);
    v[(size_t)m * D_ + n] = s;
  }
#endif
}

// ---------------------------------------------------------------------------
// K2: one block per (b, L-chunk). Single streaming pass over sequence:
//   energies -> local max/expsum -> locally-weighted partial context.
// Second chunk read hits L2 (chunk = 512 KB, L2 = 192 MB). Masked rows skipped.
// ---------------------------------------------------------------------------
__global__ __launch_bounds__(256) void k_energy_partial(const float* __restrict__ seq,
                                                        const int* __restrict__ lengths,
                                                        float* __restrict__ ws) {
  __shared__ float vsh[D_];
  __shared__ float e_sh[LC];
  __shared__ float w_sh[LC];
  __shared__ float sm_m;

  const int t  = threadIdx.x;
  const int b  = blockIdx.x / NC;
  const int c  = blockIdx.x % NC;
  const int l0 = c * LC;
  const int len = lengths[b];
  const float NEG_INF = -__builtin_inff();

  // Stage v[b,:] (4 KB) into LDS, b128 per thread.
  {
    const float4* vg = reinterpret_cast<const float4*>(ws + WS_V + (size_t)b * D_);
    reinterpret_cast<float4*>(vsh)[t] = vg[t];
  }
  __syncthreads();

  // Energies: each wave computes one dot product (D=1024) at a time.
  const int wave = t >> 5, lane = t & 31;
  const float4* vs4 = reinterpret_cast<const float4*>(vsh);
  for (int i = 0; i < LC / 8; ++i) {
    const int l = l0 + i * 8 + wave;
    float e;
    if (l < len) {
      const float4* sp = reinterpret_cast<const float4*>(seq + ((size_t)b * L_ + l) * D_);
      float s0 = 0.f;
      #pragma unroll
      for (int k = 0; k < 8; ++k) {
        const float4 x = sp[lane + 32 * k];
        const float4 y = vs4[lane + 32 * k];
        s0 = fmaf(x.x, y.x, s0); s0 = fmaf(x.y, y.y, s0);
        s0 = fmaf(x.z, y.z, s0); s0 = fmaf(x.w, y.w, s0);
      }
      #pragma unroll
      for (int off = 16; off; off >>= 1) s0 += __shfl_down(s0, off, 32);
      e = s0;
    } else {
      e = NEG_INF;  // masked position
    }
    if (lane == 0) {
      e_sh[i * 8 + wave] = e;
      ws[WS_E + (size_t)b * L_ + l] = e;
    }
  }
  __syncthreads();

  // Chunk-local max and exp-sum (wave 0).
  if (t < 32) {
    float mx = NEG_INF;
    #pragma unroll
    for (int i = 0; i < LC / 32; ++i) mx = fmaxf(mx, e_sh[t + 32 * i]);
    #pragma unroll
    for (int off = 16; off; off >>= 1) mx = fmaxf(mx, __shfl_xor(mx, off, 32));
    float sl = 0.f;
    #pragma unroll
    for (int i = 0; i < LC / 32; ++i) {
      const float e = e_sh[t + 32 * i];
      if (e > NEG_INF) sl += __expf(e - mx);
    }
    #pragma unroll
    for (int off = 16; off; off >>= 1) sl += __shfl_xor(sl, off, 32);
    if (t == 0) {
      sm_m = mx;
      ws[WS_M + b * NC + c] = mx;
      ws[WS_S + b * NC + c] = sl;
    }
  }
  __syncthreads();
  if (t < LC) {
    const float e = e_sh[t];
    w_sh[t] = (e > NEG_INF) ? __expf(e - sm_m) : 0.f;
  }
  __syncthreads();

  // Partial context: thread t owns d = 4t..4t+3; re-read chunk (L2-hot).
  float4 acc = {0.f, 0.f, 0.f, 0.f};
  const float4* sp = reinterpret_cast<const float4*>(seq + ((size_t)b * L_ + l0) * D_);
  for (int l = 0; l < LC; ++l) {
    const float w = w_sh[l];                // LDS broadcast; uniform branch
    if (w != 0.f) {
      const float4 x = sp[(size_t)l * (D_ / 4) + t];
      acc.x = fmaf(w, x.x, acc.x); acc.y = fmaf(w, x.y, acc.y);
      acc.z = fmaf(w, x.z, acc.z); acc.w = fmaf(w, x.w, acc.w);
    }
  }
  reinterpret_cast<float4*>(ws + WS_CTX + (size_t)(b * NC + c) * D_)[t] = acc;
}

// ---------------------------------------------------------------------------
// K3: per batch row, combine chunk partials (global max/sum), rescale partial
// contexts, and emit scores. Outputs: contexts [B,D] then scores [B,L].
// ---------------------------------------------------------------------------
__global__ __launch_bounds__(256) void k_combine(const float* __restrict__ ws,
                                                 float* __restrict__ out) {
  const int t = threadIdx.x;
  const int b = blockIdx.x;
  const float NEG_INF = -__builtin_inff();

  float mc[NC], sc[NC], wc[NC];
  float M = NEG_INF;
  #pragma unroll
  for (int c = 0; c < NC; ++c) {
    mc[c] = ws[WS_M + b * NC + c];
    sc[c] = ws[WS_S + b * NC + c];
    M = fmaxf(M, mc[c]);
  }
  float S = 0.f;
  #pragma unroll
  for (int c = 0; c < NC; ++c) {
    const float w = (mc[c] > NEG_INF) ? __expf(mc[c] - M) : 0.f;
    wc[c] = w;
    S += w * sc[c];
  }
  const float invS = 1.f / S;  // length >= 1 guarantees S > 0

  // Contexts
  float4 acc = {0.f, 0.f, 0.f, 0.f};
  #pragma unroll
  for (int c = 0; c < NC; ++c) {
    if (wc[c] != 0.f) {
      const float4 x =
          reinterpret_cast<const float4*>(ws + WS_CTX + (size_t)(b * NC + c) * D_)[t];
      const float w = wc[c];
      acc.x = fmaf(w, x.x, acc.x); acc.y = fmaf(w, x.y, acc.y);
      acc.z = fmaf(w, x.z, acc.z); acc.w = fmaf(w, x.w, acc.w);
    }
  }
  acc.x *= invS; acc.y *= invS; acc.z *= invS; acc.w *= invS;
  reinterpret_cast<float4*>(out + (size_t)b * D_)[t] = acc;

  // Scores
  float* sout = out + (size_t)B_ * D_ + (size_t)b * L_;
  const float* eptr = ws + WS_E + (size_t)b * L_;
  for (int l = t; l < L_; l += 256) {
    const float e = eptr[l];
    sout[l] = (e > NEG_INF) ? __expf(e - M) * invS : 0.f;
  }
}

// ---------------------------------------------------------------------------
extern "C" void kernel_launch(void* const* d_in, const int* in_sizes, int n_in,
                              void* d_out, int out_size, void* d_ws, size_t ws_size,
                              hipStream_t stream) {
  const float* seq     = (const float*)d_in[0];  // [B,L,D] f32
  const float* query   = (const float*)d_in[1];  // [B,C]   f32
  const int*   lengths = (const int*)d_in[2];    // [B]     i32
  const float* W_h     = (const float*)d_in[3];  // [C,D]   f32
  // d_in[4] = b_h: a per-row constant shift of the energies -> cancels in
  // softmax exactly; unused.
  float* out = (float*)d_out;  // contexts [B,D] ++ scores [B,L]
  float* ws  = (float*)d_ws;

  k_qw_gemm<<<dim3(128), dim3(32), 0, stream>>>(query, W_h, ws + WS_V);
  k_energy_partial<<<dim3(B_ * NC), dim3(256), 0, stream>>>(seq, lengths, ws);
  k_combine<<<dim3(B_), dim3(256), 0, stream>>>(ws, out);
}